// SoftDTW_38431367365094
// MI455X (gfx1250) — compile-verified
//
#include <hip/hip_runtime.h>

typedef float v2f __attribute__((ext_vector_type(2)));
typedef float v8f __attribute__((ext_vector_type(8)));

#define BATCH 32
#define NSEQ  512
#define DDIM  64
#define BIGF  1e30f

// ---------------------------------------------------------------------------
// Kernel 1: D[b,m,n] = |x_m|^2 + |y_n|^2 - 2 * (x_m . y_n)
// One wave32 per 16x16 tile, V_WMMA_F32_16X16X4_F32, K-loop of 16 steps.
// A (16x4 f32) layout: lane L -> row L&15, K pair 2*(L>>4)+{0,1}  (float2 load)
// B (4x16 f32) layout: lane L -> col L&15, K pair 2*(L>>4)+{0,1}  (float2 load,
//   B[k][n] = y[n][k] so it is the same row-major access as A, from y)
// C (16x16 f32) layout: VGPR v, lane L -> M = v + 8*(L>>4), N = L&15
// ---------------------------------------------------------------------------
__global__ __launch_bounds__(256) void sqdist_wmma(const float* __restrict__ X,
                                                   const float* __restrict__ Y,
                                                   float* __restrict__ Dout) {
  const int lane = threadIdx.x & 31;
  const int wave = threadIdx.x >> 5;
  const int tile = blockIdx.x * 8 + wave;      // 32*32*32 = 32768 tiles
  const int b    = tile >> 10;
  const int tm   = (tile >> 5) & 31;
  const int tn   = tile & 31;
  const int r    = lane & 15;
  const int kh   = lane >> 4;                  // which K half-pair this lane owns

  const float* xp = X + ((size_t)b * NSEQ + (size_t)tm * 16 + r) * DDIM;
  const float* yp = Y + ((size_t)b * NSEQ + (size_t)tn * 16 + r) * DDIM;

  v8f  c  = {};
  float xn = 0.0f, yn = 0.0f;                  // partial row norms (half per lane)
#pragma unroll
  for (int k0 = 0; k0 < DDIM; k0 += 4) {
    v2f a  = *(const v2f*)(xp + k0 + 2 * kh);
    v2f bm = *(const v2f*)(yp + k0 + 2 * kh);
    xn += a.x * a.x + a.y * a.y;
    yn += bm.x * bm.x + bm.y * bm.y;
    // (neg_a, A, neg_b, B, c_mod, C, reuse_a, reuse_b)
    c = __builtin_amdgcn_wmma_f32_16x16x4_f32(false, a, false, bm,
                                              (short)0, c, false, false);
  }

  // full |y|^2 for column N = r : halves live in lanes r and r+16
  float y2n = __shfl(yn, r, 32) + __shfl(yn, r + 16, 32);

  float* dcol = Dout + (size_t)b * NSEQ * NSEQ + (size_t)(tn * 16 + r);
#pragma unroll
  for (int v = 0; v < 8; ++v) {
    const int m = v + 8 * kh;                  // row of this C element
    float x2m = __shfl(xn, m, 32) + __shfl(xn, m + 16, 32);
    dcol[(size_t)(tm * 16 + m) * NSEQ] = x2m + y2n - 2.0f * c[v];
  }
}

// ---------------------------------------------------------------------------
// Kernel 2: soft-DTW anti-diagonal recursion, gamma = 1.
// One block per batch; thread tid owns row i = tid+1. Diagonals R[i, t-i]
// live in LDS, triple-buffered; one barrier per anti-diagonal step.
// Matches the reference exactly, including BIG=1e30 padding semantics.
// ---------------------------------------------------------------------------
__global__ __launch_bounds__(512) void softdtw_diag(const float* __restrict__ Dmat,
                                                    float* __restrict__ out) {
  __shared__ float buf[3][NSEQ + 1];
  const int b   = blockIdx.x;
  const int tid = threadIdx.x;
  const int i   = tid + 1;
  const float* Drow = Dmat + ((size_t)b * NSEQ + (size_t)(i - 1)) * NSEQ;

  // t=0 diagonal: R[0,0]=0, rest BIG.  t=1 diagonal: all BIG.
  for (int k = tid; k <= NSEQ; k += 512) {
    buf[0][k] = (k == 0) ? 0.0f : BIGF;
    buf[1][k] = BIGF;
  }
  __syncthreads();

  for (int t = 2; t <= 2 * NSEQ; ++t) {
    float*       cur = buf[t % 3];
    const float* p1  = buf[(t + 2) % 3];      // diagonal t-1
    const float* p2  = buf[(t + 1) % 3];      // diagonal t-2

    const int j = t - i;
    float val = BIGF;
    if (j >= 1 && j <= NSEQ) {
      float r00 = p2[i - 1];                  // R[i-1, j-1]
      float r01 = p1[i - 1];                  // R[i-1, j  ]
      float r10 = p1[i];                      // R[i,   j-1]
      float m = fminf(fminf(r00, r01), r10);
      float s = __expf(m - r00) + __expf(m - r01) + __expf(m - r10);
      val = Drow[j - 1] + (m - __logf(s));    // softmin = -log(sum exp(-r)) = m - log s
    }
    cur[i] = val;
    if (tid == 0) cur[0] = BIGF;              // R[0, t] = BIG for t >= 1
    __syncthreads();
  }

  if (tid == 0) out[b] = buf[(2 * NSEQ) % 3][NSEQ];   // R[N, M]
}

extern "C" void kernel_launch(void* const* d_in, const int* in_sizes, int n_in,
                              void* d_out, int out_size, void* d_ws, size_t ws_size,
                              hipStream_t stream) {
  const float* x = (const float*)d_in[0];     // (32, 512, 64) fp32
  const float* y = (const float*)d_in[1];     // (32, 512, 64) fp32
  float* Dws = (float*)d_ws;                  // 32*512*512*4 = 33.5 MB scratch (L2-resident)

  // Phase 1: batched pairwise sq-distance via fp32 WMMA (32768 tiles, 8 waves/block)
  sqdist_wmma<<<4096, 256, 0, stream>>>(x, y, Dws);

  // Phase 2: per-batch soft-DTW wavefront (32 blocks x 16 waves)
  softdtw_diag<<<BATCH, 512, 0, stream>>>(Dws, (float*)d_out);
}